// Spline_36996848288034
// MI455X (gfx1250) — compile-verified
//
#include <hip/hip_runtime.h>
#include <hip/hip_bf16.h>

typedef __attribute__((ext_vector_type(2))) float v2f;
typedef __attribute__((ext_vector_type(8))) float v8f;

#define WAVES_PER_BLOCK 8
#define ROWS_PER_WG (WAVES_PER_BLOCK * 16)   // 128 rows per workgroup tile
#define WP_STRIDE 80                         // float2 row stride (== 160 dwords ≡ 32 mod 64)

// ---------------------------------------------------------------- zero
__global__ void k_zero(float* p, int n) {
    int i = blockIdx.x * blockDim.x + threadIdx.x;
    if (i < n) p[i] = 0.0f;
}

// ------------------------------------------------- basis + histograms
__global__ void k_basis(const float* __restrict__ pseudo,
                        const int*   __restrict__ ei,
                        float* __restrict__ basisArr,
                        int*   __restrict__ widxArr,
                        int*   __restrict__ counts,
                        float* __restrict__ deg,
                        int E_) {
    int e = blockIdx.x * blockDim.x + threadIdx.x;
    if (e >= E_) return;
    // mul = KS - DEGREE = 4 for all dims (IS_OPEN)
    float v0 = pseudo[e * 3 + 0] * 4.0f;
    float v1 = pseudo[e * 3 + 1] * 4.0f;
    float v2 = pseudo[e * 3 + 2] * 4.0f;
    float l0 = floorf(v0), l1 = floorf(v1), l2 = floorf(v2);
    float f0 = v0 - l0, f1 = v1 - l1, f2 = v2 - l2;
    int i0 = (int)l0, i1 = (int)l1, i2 = (int)l2;

    atomicAdd(&deg[ei[e]], 1.0f);

    float bas[8];
    int   wi[8];
    #pragma unroll
    for (int s = 0; s < 8; ++s) {
        int b0 = s & 1, b1 = (s >> 1) & 1, b2 = (s >> 2) & 1;
        bas[s] = (b0 ? f0 : 1.0f - f0) *
                 (b1 ? f1 : 1.0f - f1) *
                 (b2 ? f2 : 1.0f - f2);
        wi[s] = (i0 + b0) + 5 * (i1 + b1) + 25 * (i2 + b2);  // OFFSETS = [1,5,25]
        atomicAdd(&counts[wi[s]], 1);
    }
    // vectorized stores (e*8 is 32B-aligned)
    float4* bo = (float4*)(basisArr + e * 8);
    int4*   wo = (int4*)(widxArr + e * 8);
    bo[0] = make_float4(bas[0], bas[1], bas[2], bas[3]);
    bo[1] = make_float4(bas[4], bas[5], bas[6], bas[7]);
    wo[0] = make_int4(wi[0], wi[1], wi[2], wi[3]);
    wo[1] = make_int4(wi[4], wi[5], wi[6], wi[7]);
}

// ---------------------------------------------------- tiny serial scan
__global__ void k_scan(const int* __restrict__ counts,
                       int* __restrict__ rowStart,
                       int* __restrict__ cursors,
                       int* __restrict__ tileOff) {
    if (blockIdx.x == 0 && threadIdx.x == 0) {
        int rs = 0, ts = 0;
        for (int k = 0; k < 125; ++k) {
            rowStart[k] = rs;
            cursors[k]  = rs;
            tileOff[k]  = ts;
            int c = counts[k];
            rs += c;
            ts += (c + ROWS_PER_WG - 1) / ROWS_PER_WG;
        }
        rowStart[125] = rs;
        tileOff[125]  = ts;   // total number of GEMM tiles
    }
}

// -------------------------------------------------- scatter into bins
__global__ void k_scatter(const int*   __restrict__ ei,
                          const float* __restrict__ basisArr,
                          const int*   __restrict__ widxArr,
                          int*   __restrict__ cursors,
                          int*   __restrict__ srow,
                          int*   __restrict__ scol,
                          float* __restrict__ sbas,
                          int E_) {
    int i = blockIdx.x * blockDim.x + threadIdx.x;
    if (i >= E_ * 8) return;
    int e = i >> 3;
    int k = widxArr[i];
    int pos = atomicAdd(&cursors[k], 1);
    srow[pos] = ei[e];
    scol[pos] = ei[E_ + e];
    sbas[pos] = basisArr[i];
}

// --------------------------------- binned spline GEMM (WMMA f32 16x16x4)
__global__ __launch_bounds__(256) void k_spline_gemm(
    const float* __restrict__ x,
    const float* __restrict__ weight,
    const int*   __restrict__ rowStart,
    const int*   __restrict__ tileOff,
    const int*   __restrict__ srow,
    const int*   __restrict__ scol,
    const float* __restrict__ sbas,
    float* __restrict__ msgsum) {
    // B tile pre-packed in fragment order: sWp[p][n] = {W[2p][n], W[2p+1][n]}
    __shared__ v2f   sWp[32 * WP_STRIDE];               // 20480 B, conflict-free b64 reads
    __shared__ float sA[WAVES_PER_BLOCK * 16 * 68];     // per-wave A tile, stride 68

    const int tid  = threadIdx.x;
    const int wave = tid >> 5;
    const int lane = tid & 31;
    const int half = lane >> 4;
    const int l16  = lane & 15;

    const int totalTiles = tileOff[125];

    for (int t = blockIdx.x; t < totalTiles; t += gridDim.x) {
        // binary search: largest k with tileOff[k] <= t
        int lo = 0, hi = 124;
        while (lo < hi) {
            int mid = (lo + hi + 1) >> 1;
            if (tileOff[mid] <= t) lo = mid; else hi = mid - 1;
        }
        const int k = lo;
        const int rowBase = rowStart[k] + (t - tileOff[k]) * ROWS_PER_WG;
        const int rowEnd  = rowStart[k + 1];

        // stage W_k (64x64) into LDS in packed-pair fragment order
        const float* Wk = weight + (size_t)k * 64 * 64;
        float* sWraw = (float*)sWp;
        for (int i = tid; i < 64 * 64; i += 256) {
            int r = i >> 6, c = i & 63;
            sWraw[((r >> 1) * WP_STRIDE + c) * 2 + (r & 1)] = Wk[i];
        }

        // stage this wave's 16x64 A tile (basis-scaled gathered x rows).
        // Clamp to last valid row: duplicate M-rows are independent in the
        // WMMA M dimension and the epilogue guard never stores them.
        float* A = sA + wave * 16 * 68;
        const int gr  = rowBase + wave * 16 + l16;
        const int grc = gr < rowEnd ? gr : rowEnd - 1;
        const int   colId = scol[grc];
        const float bas   = sbas[grc];
        const float4* xr4 = (const float4*)(x + (size_t)colId * 64 + half * 32);
        #pragma unroll
        for (int c = 0; c < 8; ++c) {
            float4 v = xr4[c];
            v.x *= bas; v.y *= bas; v.z *= bas; v.w *= bas;
            *(float4*)&A[l16 * 68 + half * 32 + c * 4] = v;
        }

        __syncthreads();

        v8f acc[4] = {v8f{}, v8f{}, v8f{}, v8f{}};
        #pragma unroll
        for (int kk = 0; kk < 16; ++kk) {
            const int kb = kk * 4 + 2 * half;           // fragment K rows for this half
            v2f a;
            a.x = A[l16 * 68 + kb + 0];
            a.y = A[l16 * 68 + kb + 1];
            const int bp = (kb >> 1) * WP_STRIDE + l16; // packed B base for this half
            #pragma unroll
            for (int nt = 0; nt < 4; ++nt) {
                v2f b = sWp[bp + nt * 16];
                acc[nt] = __builtin_amdgcn_wmma_f32_16x16x4_f32(
                    false, a, false, b, (short)0, acc[nt], false, false);
            }
        }

        // epilogue: atomic scatter-add into msgsum[row]
        #pragma unroll
        for (int r = 0; r < 8; ++r) {
            int mm  = r + 8 * half;
            int grr = rowBase + wave * 16 + mm;
            if (grr < rowEnd) {
                int outRow = srow[grr];
                float* dst = &msgsum[(size_t)outRow * 64 + l16];
                #pragma unroll
                for (int nt = 0; nt < 4; ++nt)
                    atomicAdd(dst + nt * 16, acc[nt][r]);
            }
        }
        __syncthreads();   // all waves done reading LDS before next tile overwrites
    }
}

// -------------------- finalize: out = msg/deg + x@root + bias (WMMA)
__global__ __launch_bounds__(256) void k_finalize(
    const float* __restrict__ x,
    const float* __restrict__ root,
    const float* __restrict__ bias,
    const float* __restrict__ msgsum,
    const float* __restrict__ deg,
    float* __restrict__ out,
    int N_) {
    __shared__ v2f   sWp[32 * WP_STRIDE];
    __shared__ float sA[WAVES_PER_BLOCK * 16 * 68];
    __shared__ float sBias[64];

    const int tid  = threadIdx.x;
    const int wave = tid >> 5;
    const int lane = tid & 31;
    const int half = lane >> 4;
    const int l16  = lane & 15;

    float* sWraw = (float*)sWp;
    for (int i = tid; i < 64 * 64; i += 256) {
        int r = i >> 6, c = i & 63;
        sWraw[((r >> 1) * WP_STRIDE + c) * 2 + (r & 1)] = root[i];
    }
    if (tid < 64) sBias[tid] = bias[tid];

    const int nodeBase = blockIdx.x * ROWS_PER_WG;
    float* A = sA + wave * 16 * 68;
    const int n  = nodeBase + wave * 16 + l16;
    const int nc = n < N_ ? n : N_ - 1;    // clamp; extra M-rows never stored
    const float4* xr4 = (const float4*)(x + (size_t)nc * 64 + half * 32);
    #pragma unroll
    for (int c = 0; c < 8; ++c)
        *(float4*)&A[l16 * 68 + half * 32 + c * 4] = xr4[c];

    __syncthreads();

    v8f acc[4] = {v8f{}, v8f{}, v8f{}, v8f{}};
    #pragma unroll
    for (int kk = 0; kk < 16; ++kk) {
        const int kb = kk * 4 + 2 * half;
        v2f a;
        a.x = A[l16 * 68 + kb + 0];
        a.y = A[l16 * 68 + kb + 1];
        const int bp = (kb >> 1) * WP_STRIDE + l16;
        #pragma unroll
        for (int nt = 0; nt < 4; ++nt) {
            v2f b = sWp[bp + nt * 16];
            acc[nt] = __builtin_amdgcn_wmma_f32_16x16x4_f32(
                false, a, false, b, (short)0, acc[nt], false, false);
        }
    }

    #pragma unroll
    for (int r = 0; r < 8; ++r) {
        int nn = nodeBase + wave * 16 + r + 8 * half;
        if (nn < N_) {
            float d = deg[nn];
            d = d < 1.0f ? 1.0f : d;
            float dinv = 1.0f / d;
            #pragma unroll
            for (int nt = 0; nt < 4; ++nt) {
                int o = nt * 16 + l16;
                out[(size_t)nn * 64 + o] =
                    acc[nt][r] + msgsum[(size_t)nn * 64 + o] * dinv + sBias[o];
            }
        }
    }
}

// ------------------------------------------------------------ launcher
extern "C" void kernel_launch(void* const* d_in, const int* in_sizes, int n_in,
                              void* d_out, int out_size, void* d_ws, size_t ws_size,
                              hipStream_t stream) {
    const float* x      = (const float*)d_in[0];
    const int*   ei     = (const int*)d_in[1];     // [2,E] (row; col)
    const float* pseudo = (const float*)d_in[2];   // [E,3]
    const float* weight = (const float*)d_in[3];   // [125,64,64]
    const float* root   = (const float*)d_in[4];   // [64,64]
    const float* bias   = (const float*)d_in[5];   // [64]

    const int N_ = in_sizes[0] / 64;
    const int E_ = in_sizes[1] / 2;

    // workspace layout (32-bit words); zeroed region is contiguous at front
    float* ws       = (float*)d_ws;
    int*   counts   = (int*)ws;                         // 128
    float* msgsum   = ws + 128;                         // N*64
    float* deg      = msgsum + (size_t)N_ * 64;         // N
    int*   rowStart = (int*)(deg + N_);                 // 128
    int*   tileOff  = rowStart + 128;                   // 128
    int*   cursors  = tileOff + 128;                    // 128
    float* basisArr = (float*)(cursors + 128);          // E*8
    int*   widxArr  = (int*)(basisArr + (size_t)E_ * 8);// E*8
    int*   srow     = widxArr + (size_t)E_ * 8;         // E*8
    int*   scol     = srow + (size_t)E_ * 8;            // E*8
    float* sbas     = (float*)(scol + (size_t)E_ * 8);  // E*8

    const int zeroN = 128 + N_ * 64 + N_;
    k_zero<<<(zeroN + 255) / 256, 256, 0, stream>>>(ws, zeroN);

    k_basis<<<(E_ + 255) / 256, 256, 0, stream>>>(pseudo, ei, basisArr, widxArr,
                                                  counts, deg, E_);

    k_scan<<<1, 32, 0, stream>>>(counts, rowStart, cursors, tileOff);

    k_scatter<<<(E_ * 8 + 255) / 256, 256, 0, stream>>>(ei, basisArr, widxArr,
                                                        cursors, srow, scol, sbas, E_);

    k_spline_gemm<<<4096, 256, 0, stream>>>(x, weight, rowStart, tileOff,
                                            srow, scol, sbas, msgsum);

    k_finalize<<<(N_ + ROWS_PER_WG - 1) / ROWS_PER_WG, 256, 0, stream>>>(
        x, root, bias, msgsum, deg, (float*)d_out, N_);
}